// ProductKeyRouter_9620726743477
// MI455X (gfx1250) — compile-verified
//
#include <hip/hip_runtime.h>
#include <math.h>

// ---------------------------------------------------------------------------
// ProductKeyRouter for MI455X (gfx1250, wave32)
//   x  [8192,2048] f32, W1 [64,2048] f32, W2 [64,2048] f32
//   s1 = x@W1^T, s2 = x@W2^T          -> S = [s1|s2]  [8192,128] (workspace)
//   scores[n, i*64+j] = s1[n,i]+s2[n,j]   (128 MiB streamed output)
//   top8 + softmax via top8(s1) x top8(s2) candidate trick (64 candidates)
// Memory-bound: ~200 MiB total traffic -> ~8.6us @ 23.3 TB/s. fp32 WMMA
// (V_WMMA_F32_16X16X4_F32) keeps reference precision for free.
// ---------------------------------------------------------------------------

typedef float v2f __attribute__((ext_vector_type(2)));
typedef float v4f __attribute__((ext_vector_type(4)));
typedef float v8f __attribute__((ext_vector_type(8)));

#define N_ROWS 8192
#define DIM    2048
#define SQRTK  64
#define TOPK   8
#define NEG_INF (-__builtin_inff())

// ---------------------------------------------------------------------------
// Kernel 1: S[8192][128] = [ x@W1^T | x@W2^T ]  via fp32 WMMA 16x16x4.
// Block = 256 threads (8 waves). Block b owns rows r0 = 16*b.
// Wave w<4  -> W1 cols 16w..16w+15 ; wave w>=4 -> W2 cols 16(w-4)..
// Per-lane WMMA layout (wave32):
//   A 16x4 : lane l -> M = l&15 ; v0 = A[M][k+2h], v1 = A[M][k+2h+1], h=l>>4
//   B 4x16 : lane l -> N = l&15 ; v0 = B[k+2h][N], v1 = B[k+2h+1][N]
//   (B[k][n] = W[n][k], same addressing pattern as A)
//   C/D 16x16 f32: acc[v] = D[v + 8*h][l&15]
// ---------------------------------------------------------------------------
__global__ void __launch_bounds__(256)
pkr_proj_kernel(const float* __restrict__ x,
                const float* __restrict__ W1,
                const float* __restrict__ W2,
                float* __restrict__ S)
{
    const int lane = threadIdx.x & 31;
    const int wave = threadIdx.x >> 5;
    const int r0   = blockIdx.x * 16;

    const int mn = lane & 15;        // M for A, N for B
    const int h  = lane >> 4;        // K-half select

    const float* Wsel = (wave < 4) ? W1 : W2;
    const int    c0   = (wave < 4) ? (wave * 16) : ((wave - 4) * 16);

    const float* ap = x    + (size_t)(r0 + mn) * DIM + 2 * h;
    const float* bp = Wsel + (size_t)(c0 + mn) * DIM + 2 * h;

    v8f acc = {};
    for (int k = 0; k < DIM; k += 16) {
        v2f a0 = *(const v2f*)(ap + k);
        v2f a1 = *(const v2f*)(ap + k + 4);
        v2f a2 = *(const v2f*)(ap + k + 8);
        v2f a3 = *(const v2f*)(ap + k + 12);
        v2f b0 = *(const v2f*)(bp + k);
        v2f b1 = *(const v2f*)(bp + k + 4);
        v2f b2 = *(const v2f*)(bp + k + 8);
        v2f b3 = *(const v2f*)(bp + k + 12);
        // (neg_a, A, neg_b, B, c_mod, C, reuse_a, reuse_b)
        acc = __builtin_amdgcn_wmma_f32_16x16x4_f32(false, a0, false, b0, (short)0, acc, false, false);
        acc = __builtin_amdgcn_wmma_f32_16x16x4_f32(false, a1, false, b1, (short)0, acc, false, false);
        acc = __builtin_amdgcn_wmma_f32_16x16x4_f32(false, a2, false, b2, (short)0, acc, false, false);
        acc = __builtin_amdgcn_wmma_f32_16x16x4_f32(false, a3, false, b3, (short)0, acc, false, false);
    }

    // Scatter C tile into S: column base = 16*wave (cols 0..63 = s1, 64..127 = s2)
    const int colbase = wave * 16;
    #pragma unroll
    for (int v = 0; v < 8; ++v) {
        int row = r0 + v + 8 * h;
        S[(size_t)row * 128 + colbase + mn] = acc[v];
    }
}

// ---------------------------------------------------------------------------
// Wave-wide top-8 selection over 64 values held 2-per-lane (wave32).
// Tie-break: lower index wins (matches jax.lax.top_k on the flat axis).
// Outputs: vals[0..7] descending (uniform across lanes); lane t also keeps
// (myv,myi) for the t-th winner.
// ---------------------------------------------------------------------------
__device__ __forceinline__ void top8_select(int lane,
                                            float v0, int i0, float v1, int i1,
                                            float vals[TOPK],
                                            float& myv, int& myi)
{
    bool c0 = false, c1 = false;
    #pragma unroll
    for (int t = 0; t < TOPK; ++t) {
        float cv0 = c0 ? NEG_INF : v0;
        float cv1 = c1 ? NEG_INF : v1;
        float v; int i;
        if (cv0 >= cv1) { v = cv0; i = i0; }   // i0 < i1 always -> tie ok
        else            { v = cv1; i = i1; }
        #pragma unroll
        for (int m = 16; m >= 1; m >>= 1) {
            float ov = __shfl_xor(v, m, 32);
            int   oi = __shfl_xor(i, m, 32);
            if (ov > v || (ov == v && oi < i)) { v = ov; i = oi; }
        }
        vals[t] = v;
        if (lane == t) { myv = v; myi = i; }
        if (i == i0) c0 = true;
        if (i == i1) c1 = true;
    }
}

// ---------------------------------------------------------------------------
// Kernel 2: per row n -- stream scores[n,:] (NT stores), top-8 + softmax.
// Block = 256 threads = 8 waves, one row per wave.
// ---------------------------------------------------------------------------
__global__ void __launch_bounds__(256)
pkr_scores_topk_kernel(const float* __restrict__ S,
                       float* __restrict__ out_scores,
                       int*   __restrict__ out_idx,
                       float* __restrict__ out_probs)
{
    __shared__ __align__(16) float sl[8][128];

    const int lane = threadIdx.x & 31;
    const int wave = threadIdx.x >> 5;
    const int n    = blockIdx.x * 8 + wave;

    // Stage this row's [s1|s2] (128 f32) into LDS.
    v4f r = *(const v4f*)(S + (size_t)n * 128 + 4 * lane);
    *(v4f*)&sl[wave][4 * lane] = r;
    __syncthreads();

    // ---- Stream the 64x64 outer-sum score tile (16 KB/row, non-temporal) ----
    const int j0 = 4 * (lane & 15);          // 16 lanes cover j = 0..63
    const int hi = lane >> 4;                // split i across lane halves
    v4f s2j;
    s2j.x = sl[wave][64 + j0 + 0];
    s2j.y = sl[wave][64 + j0 + 1];
    s2j.z = sl[wave][64 + j0 + 2];
    s2j.w = sl[wave][64 + j0 + 3];

    float* srow = out_scores + (size_t)n * (SQRTK * SQRTK);
    #pragma unroll 4
    for (int it = 0; it < 32; ++it) {
        int i = 2 * it + hi;
        float s1i = sl[wave][i];             // LDS broadcast, no conflicts
        v4f val = s2j + s1i;
        __builtin_nontemporal_store(val, (v4f*)(srow + i * SQRTK + j0));
    }

    // ---- top8(s1), top8(s2): 64 values each, 2 per lane ----
    float t1vals[TOPK], t2vals[TOPK];
    float s1mv = 0.f, s2mv = 0.f; int s1mi = 0, s2mi = 0;
    top8_select(lane, sl[wave][lane],      lane,
                      sl[wave][32 + lane], 32 + lane, t1vals, s1mv, s1mi);
    top8_select(lane, sl[wave][64 + lane],      lane,
                      sl[wave][96 + lane], 32 + lane, t2vals, s2mv, s2mi);

    // ---- 64 candidates: (top8 s1) x (top8 s2), 2 per lane ----
    // candidate p: i = p>>3, j = p&7 ; flat expert id = s1_idx*64 + s2_idx
    float av0 = __shfl(s1mv, lane >> 3, 32);
    int   ai0 = __shfl(s1mi, lane >> 3, 32);
    float av1 = __shfl(s1mv, (lane >> 3) + 4, 32);
    int   ai1 = __shfl(s1mi, (lane >> 3) + 4, 32);
    float bv  = __shfl(s2mv, lane & 7, 32);
    int   bi  = __shfl(s2mi, lane & 7, 32);

    float cv0 = av0 + bv;  int ci0 = ai0 * SQRTK + bi;
    float cv1 = av1 + bv;  int ci1 = ai1 * SQRTK + bi;
    // top8_select needs i0 < i1 only for its local tie-break; expert ids are
    // unique here and cv0/cv1 belong to distinct (i,j), so ordering by the
    // flat expert id keeps jax tie semantics.
    if (ci1 < ci0) { float tv = cv0; cv0 = cv1; cv1 = tv; int ti = ci0; ci0 = ci1; ci1 = ti; }

    float topvals[TOPK];
    float myv = 0.f; int myi = 0;
    top8_select(lane, cv0, ci0, cv1, ci1, topvals, myv, myi);

    // ---- stable softmax over the 8 winners (uniform across the wave) ----
    float m = topvals[0];
    float sum = 0.f;
    #pragma unroll
    for (int t = 0; t < TOPK; ++t) sum += __expf(topvals[t] - m);
    float inv = 1.0f / sum;

    if (lane < TOPK) {
        out_idx  [(size_t)n * TOPK + lane] = myi;                  // int32 bits
        out_probs[(size_t)n * TOPK + lane] = __expf(myv - m) * inv;
    }
}

// ---------------------------------------------------------------------------
// Launch: d_out = [ topk_idx (8192*8 int32) | probs (8192*8 f32) |
//                   scores (8192*4096 f32) ]   (reference return order)
// d_ws: S intermediate, 8192*128*4 = 4 MiB.
// ---------------------------------------------------------------------------
extern "C" void kernel_launch(void* const* d_in, const int* in_sizes, int n_in,
                              void* d_out, int out_size, void* d_ws, size_t ws_size,
                              hipStream_t stream) {
    const float* x  = (const float*)d_in[0];
    const float* W1 = (const float*)d_in[1];
    const float* W2 = (const float*)d_in[2];

    int*   out_idx    = (int*)d_out;
    float* out_probs  = (float*)d_out + (size_t)N_ROWS * TOPK;
    float* out_scores = (float*)d_out + (size_t)2 * N_ROWS * TOPK;
    float* S          = (float*)d_ws;

    pkr_proj_kernel<<<N_ROWS / 16, 256, 0, stream>>>(x, W1, W2, S);
    pkr_scores_topk_kernel<<<N_ROWS / 8, 256, 0, stream>>>(S, out_scores, out_idx, out_probs);
}